// RecVAE_loss_48515950576387
// MI455X (gfx1250) — compile-verified
//
#include <hip/hip_runtime.h>

// ---------------------------------------------------------------------------
// RecVAE loss for MI455X (gfx1250).  HBM-bound streaming reduction:
//   traffic = 4 arrays * 1024*50000*4B = 800MB  ->  ~34us @ 23.3 TB/s.
// No matmul FLOPs exist, so WMMA is inapplicable; the CDNA5 feature used is
// the async global->LDS data mover (ASYNCcnt) for a zero-VGPR-cost double-
// buffered stream pipeline, per CDNA5 ISA.
// ---------------------------------------------------------------------------

typedef float v4f __attribute__((ext_vector_type(4)));
typedef int   v4i __attribute__((vector_size(16)));   // matches builtin param

#define B_ROWS    1024
#define N_ITEMS   50000
#define D_DIM     256
#define NV4       (N_ITEMS / 4)        // 12500 float4 per row
#define TPB       256
#define TILE      256                  // float4s per tile per stream
#define NTILES    (NV4 / TILE)         // 48 full tiles
#define TAILBEG   (NTILES * TILE)      // 12288
#define K2_BLOCKS 128

#define ANNEAL_F  1.0f
#define EPS_F     1.0f

#define NEG_INF   (-__builtin_inff())

// ---- CDNA5 async global->LDS copy (16B per lane) --------------------------
#if defined(__gfx1250__) && __has_builtin(__builtin_amdgcn_global_load_async_to_lds_b128)
#define USE_ASYNC_BUILTIN 1
#else
#define USE_ASYNC_BUILTIN 0
#endif

typedef __attribute__((address_space(1))) v4i as1_v4i;
typedef __attribute__((address_space(3))) v4i as3_v4i;

__device__ __forceinline__ void async_ld16(const v4f* g, v4f* l) {
#if USE_ASYNC_BUILTIN
    __builtin_amdgcn_global_load_async_to_lds_b128(
        (as1_v4i*)g,            // global source, int4 pointee per builtin sig
        (as3_v4i*)l,            // LDS destination
        /*offset=*/0, /*cpol=*/0);
#else
    unsigned lds_off =
        (unsigned)(unsigned long long)(__attribute__((address_space(3))) char*)(void*)l;
    asm volatile("global_load_async_to_lds_b128 %0, %1, off"
                 :: "v"(lds_off), "v"(g) : "memory");
#endif
}

template <int N>
__device__ __forceinline__ void wait_asynccnt() {
#if defined(__gfx1250__) && __has_builtin(__builtin_amdgcn_s_wait_asynccnt)
    __builtin_amdgcn_s_wait_asynccnt(N);
#else
    asm volatile("s_wait_asynccnt %0" :: "i"(N) : "memory");
#endif
    asm volatile("" ::: "memory");   // compiler ordering fence for LDS reads
}

// ---- online log-sum-exp + dot accumulator ---------------------------------
__device__ __forceinline__ void acc_stream(float l, float xv,
                                           float& m, float& s, float& d) {
    d = fmaf(l, xv, d);
    if (l > m) {            // rare after warm-up: ~1 exp per element on average
        s *= __expf(m - l);
        m = l;
    }
    s += __expf(l - m);
}

// ---------------------------------------------------------------------------
// Kernel 1: one workgroup per row; fused 3-stream log_softmax dot reduction.
// terms[k*B_ROWS + row] = sum_i l_i x_i - (max + log sumexp) * sum_i x_i
//   k = 0: recon_x (merged), 1: logits_text, 2: logits_rec
// ---------------------------------------------------------------------------
__global__ __launch_bounds__(TPB) void bce_rows_kernel(
    const float* __restrict__ recon_x, const float* __restrict__ x,
    const float* __restrict__ ltext,   const float* __restrict__ lrec,
    float* __restrict__ terms)
{
    __shared__ v4f tilebuf[2][4][TILE];   // 2 bufs x 4 streams x 256 x 16B = 32KB

    const int row = blockIdx.x;
    const int tid = threadIdx.x;
    const size_t base = (size_t)row * N_ITEMS;
    const v4f* gx = (const v4f*)(x       + base);
    const v4f* g0 = (const v4f*)(recon_x + base);
    const v4f* g1 = (const v4f*)(ltext   + base);
    const v4f* g2 = (const v4f*)(lrec    + base);

    float m0 = NEG_INF, s0 = 0.f, d0 = 0.f;
    float m1 = NEG_INF, s1 = 0.f, d1 = 0.f;
    float m2 = NEG_INF, s2 = 0.f, d2 = 0.f;
    float sx = 0.f;

    // prologue: prefetch tile 0 (4 async loads -> ASYNCcnt = 4)
    async_ld16(gx + tid, &tilebuf[0][0][tid]);
    async_ld16(g0 + tid, &tilebuf[0][1][tid]);
    async_ld16(g1 + tid, &tilebuf[0][2][tid]);
    async_ld16(g2 + tid, &tilebuf[0][3][tid]);

    for (int t = 0; t < NTILES; ++t) {
        const int cb = t & 1;
        if (t + 1 < NTILES) {
            const int nb = (t + 1) & 1;
            const int i  = (t + 1) * TILE + tid;
            async_ld16(gx + i, &tilebuf[nb][0][tid]);
            async_ld16(g0 + i, &tilebuf[nb][1][tid]);
            async_ld16(g1 + i, &tilebuf[nb][2][tid]);
            async_ld16(g2 + i, &tilebuf[nb][3][tid]);
            wait_asynccnt<4>();   // tile t complete (in-order), t+1 in flight
        } else {
            wait_asynccnt<0>();
        }
        // Each thread reads only LDS it wrote itself -> no barrier needed.
        v4f xv = tilebuf[cb][0][tid];
        v4f la = tilebuf[cb][1][tid];
        v4f lb = tilebuf[cb][2][tid];
        v4f lc = tilebuf[cb][3][tid];
        #pragma unroll
        for (int j = 0; j < 4; ++j) {
            float xj = xv[j];
            sx += xj;
            acc_stream(la[j], xj, m0, s0, d0);
            acc_stream(lb[j], xj, m1, s1, d1);
            acc_stream(lc[j], xj, m2, s2, d2);
        }
    }

    // tail: 12500 - 12288 = 212 float4s, direct loads
    for (int i = TAILBEG + tid; i < NV4; i += TPB) {
        v4f xv = gx[i], la = g0[i], lb = g1[i], lc = g2[i];
        #pragma unroll
        for (int j = 0; j < 4; ++j) {
            float xj = xv[j];
            sx += xj;
            acc_stream(la[j], xj, m0, s0, d0);
            acc_stream(lb[j], xj, m1, s1, d1);
            acc_stream(lc[j], xj, m2, s2, d2);
        }
    }

    // workgroup reduction (reuse tile LDS: 10 * 256 floats = 10KB)
    __syncthreads();
    float* red = (float*)&tilebuf[0][0][0];
    red[0*TPB+tid] = m0; red[1*TPB+tid] = s0; red[2*TPB+tid] = d0;
    red[3*TPB+tid] = m1; red[4*TPB+tid] = s1; red[5*TPB+tid] = d1;
    red[6*TPB+tid] = m2; red[7*TPB+tid] = s2; red[8*TPB+tid] = d2;
    red[9*TPB+tid] = sx;
    __syncthreads();
    for (int st = TPB / 2; st > 0; st >>= 1) {
        if (tid < st) {
            #pragma unroll
            for (int k = 0; k < 3; ++k) {
                float ma = red[(3*k+0)*TPB + tid], mb = red[(3*k+0)*TPB + tid + st];
                float sa = red[(3*k+1)*TPB + tid], sb = red[(3*k+1)*TPB + tid + st];
                float mm = fmaxf(ma, mb);
                red[(3*k+0)*TPB + tid] = mm;
                red[(3*k+1)*TPB + tid] = sa * __expf(ma - mm) + sb * __expf(mb - mm);
                red[(3*k+2)*TPB + tid] += red[(3*k+2)*TPB + tid + st];
            }
            red[9*TPB + tid] += red[9*TPB + tid + st];
        }
        __syncthreads();
    }
    if (tid == 0) {
        float sxt = red[9*TPB];
        #pragma unroll
        for (int k = 0; k < 3; ++k) {
            float mm = red[(3*k+0)*TPB];
            float ss = red[(3*k+1)*TPB];
            float dd = red[(3*k+2)*TPB];
            terms[k * B_ROWS + row] = dd - (mm + __logf(ss)) * sxt;
        }
    }
}

// ---------------------------------------------------------------------------
// Kernel 2: latent terms over (B, D).  Per-block partial sums of
//   KLD1 elems: 1 + lv - mu^2 - e^lv
//   KLD2 elems: -0.5*plv - 0.5*(z-pmu)^2 * e^-plv + 0.5*z^2
//   Wass elems: (mu-pmu)^2 + e^lv + e^plv - 2*e^{(lv+plv)/2}
// ---------------------------------------------------------------------------
__global__ __launch_bounds__(TPB) void latent_kernel(
    const float* __restrict__ z,   const float* __restrict__ mu,
    const float* __restrict__ lv,  const float* __restrict__ pmu,
    const float* __restrict__ plv, float* __restrict__ partials)
{
    const int tid = threadIdx.x;
    const int n = B_ROWS * D_DIM;
    float a1 = 0.f, a2 = 0.f, aw = 0.f;
    for (int i = blockIdx.x * TPB + tid; i < n; i += gridDim.x * TPB) {
        float zz = z[i], m = mu[i], l = lv[i], pm = pmu[i], pl = plv[i];
        float el = __expf(l), ep = __expf(pl);
        a1 += 1.0f + l - m * m - el;
        float dz = zz - pm;
        a2 += -0.5f * pl - 0.5f * dz * dz * __expf(-pl) + 0.5f * zz * zz;
        float dm = m - pm;
        aw += dm * dm + el + ep - 2.0f * __expf(0.5f * (l + pl));
    }
    __shared__ float r[3][TPB];
    r[0][tid] = a1; r[1][tid] = a2; r[2][tid] = aw;
    __syncthreads();
    for (int st = TPB / 2; st > 0; st >>= 1) {
        if (tid < st) {
            r[0][tid] += r[0][tid + st];
            r[1][tid] += r[1][tid + st];
            r[2][tid] += r[2][tid + st];
        }
        __syncthreads();
    }
    if (tid == 0) {
        partials[blockIdx.x * 3 + 0] = r[0][0];
        partials[blockIdx.x * 3 + 1] = r[1][0];
        partials[blockIdx.x * 3 + 2] = r[2][0];
    }
}

// ---------------------------------------------------------------------------
// Kernel 3: final reduction + scalar assembly.
// out = (l, BCE, wasserstein, BCE_rec, BCE_text, BCE_merged)
// ---------------------------------------------------------------------------
__global__ __launch_bounds__(TPB) void finalize_kernel(
    const float* __restrict__ terms, const float* __restrict__ partials,
    float* __restrict__ out)
{
    const int tid = threadIdx.x;
    float t0 = 0.f, t1 = 0.f, t2 = 0.f, p0 = 0.f, p1 = 0.f, p2 = 0.f;
    for (int i = tid; i < B_ROWS; i += TPB) {
        t0 += terms[0 * B_ROWS + i];
        t1 += terms[1 * B_ROWS + i];
        t2 += terms[2 * B_ROWS + i];
    }
    for (int i = tid; i < K2_BLOCKS; i += TPB) {
        p0 += partials[3 * i + 0];
        p1 += partials[3 * i + 1];
        p2 += partials[3 * i + 2];
    }
    __shared__ float r[6][TPB];
    r[0][tid] = t0; r[1][tid] = t1; r[2][tid] = t2;
    r[3][tid] = p0; r[4][tid] = p1; r[5][tid] = p2;
    __syncthreads();
    for (int st = TPB / 2; st > 0; st >>= 1) {
        if (tid < st) {
            #pragma unroll
            for (int q = 0; q < 6; ++q) r[q][tid] += r[q][tid + st];
        }
        __syncthreads();
    }
    if (tid == 0) {
        const float inv_bn = 1.0f / ((float)B_ROWS * (float)N_ITEMS);
        const float inv_bd = 1.0f / ((float)B_ROWS * (float)D_DIM);
        float bce_m = -r[0][0] * inv_bn;
        float bce_t = -r[1][0] * inv_bn;
        float bce_r = -r[2][0] * inv_bn;
        float bce   = (bce_m + bce_t + bce_r) * (1.0f / 3.0f);
        float kld1  = -0.5f * r[3][0] * inv_bd;
        float kld2  = ANNEAL_F * r[4][0] * inv_bd;
        float wass  = r[5][0] * (1.0f / (float)B_ROWS);
        float L     = bce + ANNEAL_F * 0.5f * (kld1 + kld2) + EPS_F * wass;
        out[0] = L;
        out[1] = bce;
        out[2] = wass;
        out[3] = bce_r;
        out[4] = bce_t;
        out[5] = bce_m;
    }
}

// ---------------------------------------------------------------------------
extern "C" void kernel_launch(void* const* d_in, const int* in_sizes, int n_in,
                              void* d_out, int out_size, void* d_ws, size_t ws_size,
                              hipStream_t stream) {
    (void)in_sizes; (void)n_in; (void)out_size; (void)ws_size;
    const float* recon_x = (const float*)d_in[0];
    const float* x       = (const float*)d_in[1];
    const float* z       = (const float*)d_in[2];
    const float* mu      = (const float*)d_in[3];
    const float* logvar  = (const float*)d_in[4];
    const float* ltext   = (const float*)d_in[5];
    const float* lrec    = (const float*)d_in[6];
    const float* pmu     = (const float*)d_in[7];
    const float* plv     = (const float*)d_in[8];
    // d_in[9] (train_items) is unused by the reference.

    float* ws       = (float*)d_ws;
    float* terms    = ws;                 // 3 * 1024 floats (fully overwritten)
    float* partials = ws + 3 * B_ROWS;    // 128 * 3 floats (fully overwritten)
    float* out      = (float*)d_out;

    bce_rows_kernel<<<B_ROWS, TPB, 0, stream>>>(recon_x, x, ltext, lrec, terms);
    latent_kernel<<<K2_BLOCKS, TPB, 0, stream>>>(z, mu, logvar, pmu, plv, partials);
    finalize_kernel<<<1, TPB, 0, stream>>>(terms, partials, out);
}